// Shift_89936615179047
// MI455X (gfx1250) — compile-verified
//
#include <hip/hip_runtime.h>

// Shift: out[c*9 + (dy*3+dx), y, x] = x[c, y+dy-1, x+dx-1] (zero-padded), C=288, H=W=256.
// Pure data movement: 75.5 MB in (read once via async global->LDS), 679 MB out (NT stores).

#define C_DIM 288
#define HW    256
#define TILE_H 8
#define LDS_ROWS   (TILE_H + 2)   // 10: rows y0-1 .. y0+8
#define LDS_STRIDE 264            // floats; 264*4 = 1056 B row stride -> every row 16B aligned
                                  // interior (global cols 0..255) at LDS cols 4..259 (+16B, 16B aligned)
                                  // col 3 = x=-1 pad (always 0), col 260 = x=256 pad (always 0)

typedef float v4f __attribute__((ext_vector_type(4)));

__global__ __launch_bounds__(256) void shift9_kernel(const float* __restrict__ x,
                                                     float* __restrict__ out) {
  __shared__ float tile[LDS_ROWS * LDS_STRIDE];  // 10,560 B

  const int tid = threadIdx.x;
  const int t   = blockIdx.x;            // row-tile index 0..31
  const int c   = blockIdx.y;            // channel 0..287
  const int y0  = t * TILE_H;

  // ---- Phase 0: zero the halo (y-halo rows 0 and 9; x-halo cols 3 and 260) ----
  for (int i = tid; i < 2 * LDS_STRIDE; i += 256) {
    const int r   = (i < LDS_STRIDE) ? 0 : (LDS_ROWS - 1);
    const int col = (i < LDS_STRIDE) ? i : (i - LDS_STRIDE);
    tile[r * LDS_STRIDE + col] = 0.0f;
  }
  if (tid < LDS_ROWS) {
    tile[tid * LDS_STRIDE + 3]   = 0.0f;   // x = -1  (image left pad)
    tile[tid * LDS_STRIDE + 260] = 0.0f;   // x = 256 (image right pad)
  }
  __syncthreads();  // zeros visible before async writes can land

  // ---- Phase 1: async global -> LDS, B128 per lane (16 B both sides, 16 B aligned) ----
  const float* xc = x + (size_t)c * HW * HW;
  for (int job = tid; job < LDS_ROWS * 64; job += 256) {   // 640 lane-transfers
    const int r   = job >> 6;        // LDS row 0..9
    const int seg = job & 63;        // 16-byte segment 0..63
    const int yg  = y0 - 1 + r;      // global row
    if (yg >= 0 && yg < HW) {        // EXEC-masked: OOB rows keep their pre-zeroed halo
      const unsigned lds_byte = (unsigned)(r * LDS_STRIDE + 4 + seg * 4) * 4u;
      const float* gp = xc + (size_t)yg * HW + seg * 4;
      asm volatile("global_load_async_to_lds_b128 %0, %1, off"
                   :
                   : "v"(lds_byte), "v"(gp)
                   : "memory");
    }
  }
  asm volatile("s_wait_asynccnt 0x0" ::: "memory");
  __syncthreads();  // all waves' async data visible to all waves

  // ---- Phase 2: 9 shifted copies from LDS, non-temporal B128 stores ----
  for (int j = 0; j < 9; ++j) {
    const int dy = j / 3;
    const int dx = j % 3;
    float* op = out + (((size_t)c * 9 + j) * HW + y0) * HW;
    for (int job = tid; job < TILE_H * 64; job += 256) {   // 512 float4 stores per shift
      const int row = job >> 6;      // 0..7
      const int seg = job & 63;      // 0..63
      // need global col x+dx-1 for x = seg*4 .. seg*4+3 -> LDS col 3 + seg*4 + dx + m
      const float* s = &tile[(row + dy) * LDS_STRIDE + 3 + seg * 4 + dx];
      v4f v = { s[0], s[1], s[2], s[3] };
      __builtin_nontemporal_store(v, (v4f*)(op + (size_t)row * HW + seg * 4));
    }
  }
}

extern "C" void kernel_launch(void* const* d_in, const int* in_sizes, int n_in,
                              void* d_out, int out_size, void* d_ws, size_t ws_size,
                              hipStream_t stream) {
  (void)in_sizes; (void)n_in; (void)d_ws; (void)ws_size; (void)out_size;
  const float* x = (const float*)d_in[0];
  float* out = (float*)d_out;
  dim3 grid(HW / TILE_H, C_DIM);   // (32, 288) blocks
  dim3 block(256);                 // 8 wave32s
  shift9_kernel<<<grid, block, 0, stream>>>(x, out);
}